// RandomPointSampling_12627203850880
// MI455X (gfx1250) — compile-verified
//
#include <hip/hip_runtime.h>

// RandomPointSampling for MI455X (gfx1250, wave32).
// d_in[0]: pred_cloud f32 (32, 524288, 3). d_out: f32 (32, 16384, 3).
//
// Design: one thread per output sample. Deterministic hash -> random start
// index; probe forward until a point with x+y+z != 0 is found (expected 1.43
// probes at 70% density). First candidate is gathered via the CDNA5 async
// global->LDS path (GLOBAL_LOAD_ASYNC_TO_LDS_B32, ASYNCcnt / s_wait_asynccnt);
// re-probes use direct loads. No atomics -> bitwise deterministic replays.

#define RPS_BATCHES   32u
#define RPS_NPTS      524288u
#define RPS_NMASK     (RPS_NPTS - 1u)
#define RPS_NSAMP     16384u
#define RPS_MAXPROBES 2048

#if defined(__HIP_DEVICE_COMPILE__) && __has_builtin(__builtin_amdgcn_global_load_async_to_lds_b32)
#define RPS_HAVE_ASYNC_LDS 1
#else
#define RPS_HAVE_ASYNC_LDS 0
#endif

// Builtin signature (from hipcc diagnostic): param 1 is AS1 ("__device__") int*;
// destination is the LDS (AS3) int*.
typedef __attribute__((address_space(1))) int rps_gint;  // global int
typedef __attribute__((address_space(3))) int rps_lint;  // LDS int

__device__ __forceinline__ unsigned rps_mix32(unsigned x) {
  // lowbias32 avalanche hash (deterministic stand-in for the reference PRNG)
  x ^= x >> 16; x *= 0x7feb352dU;
  x ^= x >> 15; x *= 0x846ca68bU;
  x ^= x >> 16;
  return x;
}

__device__ __forceinline__ void rps_wait_async0() {
#if RPS_HAVE_ASYNC_LDS
#if __has_builtin(__builtin_amdgcn_s_wait_asynccnt)
  __builtin_amdgcn_s_wait_asynccnt(0);
#else
  asm volatile("s_wait_asynccnt 0" ::: "memory");
#endif
  __asm__ __volatile__("" ::: "memory");  // keep LDS reads after the wait
#endif
}

__global__ __launch_bounds__(256) void rps_sample_kernel(
    const float* __restrict__ pts, float* __restrict__ out) {
  const unsigned tid = blockIdx.x * 256u + threadIdx.x;   // 0 .. 32*16384-1
  const unsigned b   = tid >> 14;                          // batch
  const unsigned j   = tid & (RPS_NSAMP - 1u);             // sample id
  const float* __restrict__ base =
      pts + (unsigned long long)b * (unsigned long long)(RPS_NPTS * 3u);

  unsigned h = rps_mix32((b * 0x9E3779B9u) ^ (j * 0x85EBCA6Bu) ^ 0x6C62272Eu);
  unsigned r = h & RPS_NMASK;

  float x, y, z;

#if RPS_HAVE_ASYNC_LDS
  // Stage the first candidate (12B/lane) via async global->LDS gather.
  __shared__ float stage[256 * 4];  // 16B-aligned slot per thread
  const unsigned slot = threadIdx.x * 4u;
  {
    const float* gsrc = base + (unsigned long long)r * 3u;
    unsigned long long ga = (unsigned long long)gsrc;      // flat == AS1 bits
    unsigned la = (unsigned)(unsigned long long)(stage + slot);  // addr[31:0]
    rps_gint* gp = (rps_gint*)ga;
    rps_lint* lp = (rps_lint*)la;
    // INST_OFFSET is added to BOTH the global and LDS addresses (ISA 08 §4.4).
    __builtin_amdgcn_global_load_async_to_lds_b32(gp, lp, 0, 0);
    __builtin_amdgcn_global_load_async_to_lds_b32(gp, lp, 4, 0);
    __builtin_amdgcn_global_load_async_to_lds_b32(gp, lp, 8, 0);
  }
  rps_wait_async0();
  x = stage[slot + 0];
  y = stage[slot + 1];
  z = stage[slot + 2];

  int probes = 0;
  while ((x + y + z) == 0.0f && probes < RPS_MAXPROBES) {
    r = (r + 1u) & RPS_NMASK;
    const float* p = base + (unsigned long long)r * 3u;
    x = p[0]; y = p[1]; z = p[2];
    ++probes;
  }
#else
  x = 0.0f; y = 0.0f; z = 0.0f;
  for (int probe = 0; probe <= RPS_MAXPROBES; ++probe) {
    const float* p = base + (unsigned long long)r * 3u;
    x = p[0]; y = p[1]; z = p[2];
    if ((x + y + z) != 0.0f) break;
    r = (r + 1u) & RPS_NMASK;
  }
#endif

  float* o = out + (unsigned long long)tid * 3u;
  o[0] = x; o[1] = y; o[2] = z;
}

extern "C" void kernel_launch(void* const* d_in, const int* in_sizes, int n_in,
                              void* d_out, int out_size, void* d_ws,
                              size_t ws_size, hipStream_t stream) {
  (void)in_sizes; (void)n_in; (void)out_size; (void)d_ws; (void)ws_size;
  const float* pts = (const float*)d_in[0];
  float* out = (float*)d_out;
  const unsigned total = RPS_BATCHES * RPS_NSAMP;  // 524288 threads
  dim3 grid(total / 256u), block(256u);
  hipLaunchKernelGGL(rps_sample_kernel, grid, block, 0, stream, pts, out);
}